// TransformambaBlock_88167088652788
// MI455X (gfx1250) — compile-verified
//
#include <hip/hip_runtime.h>
#include <cstddef>

// ---------------------------------------------------------------------------
// TransformambaBlock for MI455X (gfx1250, wave32, WMMA).
// All GEMMs use V_WMMA_F32_16X16X4_F32 (f32 A/B, f32 accum): the block is
// memory/latency bound (~8 GFLOP vs multi-PFLOP matrix ceiling), so we keep
// f32 precision for free. Attention is flash-style (no NxN materialization).
// GEMM is templated on BOUNDED: the common (full-tile) instantiation has a
// pure load+WMMA inner loop; the bounded one uses clamped addresses + 0/1
// mask multiplies so EXEC stays all-ones at every WMMA (no saveexec churn).
// ---------------------------------------------------------------------------

#define B_   8
#define N_   1024
#define D_   64
#define H_   4
#define DH_  16
#define DI_  128
#define DS_  64
#define ROWS (B_ * N_)      // 8192
#define TFF  1027           // c1 conv output length (N + 3)
#define TPAD 1030           // padded time length for c1 input

typedef __attribute__((ext_vector_type(2))) float v2f;
typedef __attribute__((ext_vector_type(8))) float v8f;

__device__ inline v8f zero8() {
  v8f v;
  for (int i = 0; i < 8; ++i) v[i] = 0.0f;
  return v;
}

__device__ inline v8f wmma_f32(v2f a, v2f b, v8f c) {
  // D = A(16x4) * B(4x16) + C(16x16), f32 everywhere.
  return __builtin_amdgcn_wmma_f32_16x16x4_f32(false, a, false, b, (short)0, c,
                                               false, false);
}

__device__ inline float sigmoidf_(float x) { return 1.0f / (1.0f + __expf(-x)); }
__device__ inline float siluf_(float x) { return x * sigmoidf_(x); }

// ---------------------------------------------------------------------------
// LayerNorm over 64 features, one wave per row (lane holds cols lane, lane+32)
// ---------------------------------------------------------------------------
__global__ void ln64_kernel(const float* __restrict__ in,
                            const float* __restrict__ gw,
                            const float* __restrict__ gb,
                            float* __restrict__ out, int nrows) {
  int wv = (int)((blockIdx.x * blockDim.x + threadIdx.x) >> 5);
  int lane = threadIdx.x & 31;
  if (wv >= nrows) return;
  const float* r = in + (size_t)wv * D_;
  float x0 = r[lane], x1 = r[lane + 32];
  float s = x0 + x1;
  for (int m = 16; m >= 1; m >>= 1) s += __shfl_xor(s, m, 32);
  float mean = s * (1.0f / 64.0f);
  float d0 = x0 - mean, d1 = x1 - mean;
  float v = d0 * d0 + d1 * d1;
  for (int m = 16; m >= 1; m >>= 1) v += __shfl_xor(v, m, 32);
  float inv = rsqrtf(v * (1.0f / 64.0f) + 1e-5f);
  float* o = out + (size_t)wv * D_;
  o[lane]      = d0 * inv * gw[lane] + gb[lane];
  o[lane + 32] = d1 * inv * gw[lane + 32] + gb[lane + 32];
}

// ---------------------------------------------------------------------------
// Generic 16x16-tile WMMA GEMM:  C[b] = act(A[b] @ B + bias) + resScale*res[b]
// 8 waves per block; wave w owns M-tile (blockIdx.y*8 + w). A physical row =
// out_row + aRowOff (shifted-pointer conv-as-GEMM). B row-major KxN, shared.
// act: 0 none, 1 softplus, 2 leaky(0.01). accumulate: C += instead of C =.
// BOUNDED=false requires M % (8*16) == 0 and N % 16 == 0: inner loop is pure
// loads + WMMA. BOUNDED=true clamps addresses and masks with 0/1 multiplies.
// K must be a multiple of 4; lda must be even (8B-aligned v2f A loads).
// ---------------------------------------------------------------------------
template <bool BOUNDED>
__global__ __launch_bounds__(256)
void gemm16_wmma(const float* __restrict__ A, const float* __restrict__ Bm,
                 float* __restrict__ C, int M, int N, int K, int lda,
                 size_t aBatch, int aRowOff, int ldc, size_t cBatch,
                 const float* __restrict__ bias, const float* __restrict__ res,
                 float resScale, int act, int accumulate) {
  int wv = threadIdx.x >> 5;
  int lane = threadIdx.x & 31;
  int col0 = blockIdx.x * 16;
  int row0 = (blockIdx.y * 8 + wv) * 16;
  int bb = blockIdx.z;
  const float* Ab = A + (size_t)bb * aBatch;
  float* Cb = C + (size_t)bb * cBatch;
  const float* Rb = res ? (res + (size_t)bb * cBatch) : (const float*)0;

  int mrow = lane & 15;            // A row within tile / B,C col within tile
  int kb = (lane >> 4) << 1;       // K sub-offset: lanes 0-15 -> 0,1; 16-31 -> 2,3
  int rbase = (lane >> 4) << 3;    // C rows: lanes 0-15 -> 0..7; 16-31 -> 8..15

  int ar = row0 + mrow;
  int bc = col0 + mrow;
  float amask = 1.0f, bmask = 1.0f;
  int arc = ar, bcc = bc;
  if (BOUNDED) {
    amask = (ar < M) ? 1.0f : 0.0f;
    bmask = (bc < N) ? 1.0f : 0.0f;
    arc = (ar < M) ? ar : (M - 1);          // clamp: always in-bounds
    bcc = (bc < N) ? bc : (N - 1);
  }

  const float* ap = Ab + (size_t)(arc + aRowOff) * lda + kb;   // 8B aligned
  const float* bp0 = Bm + (size_t)kb * N + bcc;                // row kb
  const float* bp1 = bp0 + N;                                  // row kb+1
  size_t bstep = (size_t)4 * N;

  v8f acc = zero8();
#pragma unroll 4
  for (int k = 0; k < K; k += 4) {
    v2f a = *(const v2f*)ap;
    v2f b;
    b.x = bp0[0];
    b.y = bp1[0];
    if (BOUNDED) {
      a.x *= amask;
      a.y *= amask;
      b.x *= bmask;
      b.y *= bmask;
    }
    acc = wmma_f32(a, b, acc);
    ap += 4;
    bp0 += bstep;
    bp1 += bstep;
  }

  if (!BOUNDED || bc < N) {
    float bv = bias ? bias[bc] : 0.0f;
    for (int r = 0; r < 8; ++r) {
      int rr = row0 + rbase + r;
      if (!BOUNDED || rr < M) {
        float vout = acc[r] + bv;
        if (act == 1) vout = (vout > 20.0f) ? vout : logf(1.0f + __expf(vout));
        else if (act == 2) vout = (vout >= 0.0f) ? vout : 0.01f * vout;
        if (Rb) vout += resScale * Rb[(size_t)rr * ldc + bc];
        float* cp = Cb + (size_t)rr * ldc + bc;
        if (accumulate) *cp += vout; else *cp = vout;
      }
    }
  }
}

// ---------------------------------------------------------------------------
// Flash attention. 4 waves per block; each wave owns one (b, h, 16-row query
// tile) with private LDS slabs. Streams over 64 key tiles: S = Q@K^T (WMMA) +
// rel-pos bias (VALU, rel rows staged in LDS) + drop-mask, online softmax,
// O += P@V (P staged via LDS for A-layout). Trip counts are uniform, so the
// block-wide barriers pair up across waves.
// ---------------------------------------------------------------------------
__global__ __launch_bounds__(128)
void attn_kernel(const float* __restrict__ q, const float* __restrict__ kv,
                 const float* __restrict__ rel,
                 const unsigned char* __restrict__ mask,
                 float* __restrict__ obuf) {
  int wv = threadIdx.x >> 5;
  int lane = threadIdx.x & 31;
  int blk = blockIdx.x;                 // [0, B*H*16): 4 q-tiles per block
  int qt = (blk & 15) * 4 + wv;
  int h = (blk >> 4) & 3;
  int b = blk >> 6;
  int i0 = qt * 16;

  int mrow = lane & 15;
  int kb = (lane >> 4) << 1;
  int rbase = (lane >> 4) << 3;

  __shared__ float sQ[4][16][16];
  __shared__ float sP[4][16][16];
  __shared__ float sR[4][31][16];

  for (int idx = lane; idx < 256; idx += 32) {
    int r = idx >> 4, d = idx & 15;
    sQ[wv][r][d] = q[((size_t)b * N_ + i0 + r) * D_ + h * DH_ + d];
  }
  __syncthreads();

  v2f aq[4];
  for (int kc = 0; kc < 4; ++kc) {
    aq[kc].x = sQ[wv][mrow][kc * 4 + kb];
    aq[kc].y = sQ[wv][mrow][kc * 4 + kb + 1];
  }

  float mrun[8], lrun[8];
  for (int r = 0; r < 8; ++r) { mrun[r] = -1e30f; lrun[r] = 0.0f; }
  v8f oacc = zero8();
  const float scale = 0.25f;  // DH^-0.5

  for (int j0 = 0; j0 < N_; j0 += 16) {
    __syncthreads();  // previous iteration's sP/sR reads complete
    // rel-pos rows for this tile: i-j in [i0-j0-15, i0-j0+15]
    for (int idx = lane; idx < 31 * 16; idx += 32) {
      int t = idx >> 4, d = idx & 15;
      int dist = i0 - j0 - 15 + t;
      dist = dist < -512 ? -512 : (dist > 512 ? 512 : dist);
      sR[wv][t][d] = rel[(size_t)(dist + 512) * DH_ + d];
    }
    __syncthreads();

    // S tile = Q @ K^T  (contraction over DH=16 -> 4 WMMAs)
    v8f s = zero8();
    for (int kc = 0; kc < 4; ++kc) {
      v2f bf;
      size_t kr = ((size_t)b * N_ + j0 + mrow) * (2 * D_) + h * DH_ + kc * 4;
      bf.x = kv[kr + kb];
      bf.y = kv[kr + kb + 1];
      s = wmma_f32(aq[kc], bf, s);
    }

    // per-element: +rel bias, scale, mask; online softmax
    float sv[8];
    for (int r = 0; r < 8; ++r) {
      int row = rbase + r;
      float rp = 0.0f;
      int t = row - mrow + 15;
      for (int d = 0; d < 16; ++d) rp += sQ[wv][row][d] * sR[wv][t][d];
      float val = (s[r] + rp) * scale;
      size_t mi = ((((size_t)(b * H_ + h)) * N_ + (i0 + row)) << 10) + j0 + mrow;
      val += mask[mi] ? -1e12f : 0.0f;
      sv[r] = val;
    }
    for (int r = 0; r < 8; ++r) {
      float tmax = sv[r];
      for (int xm = 1; xm < 16; xm <<= 1)
        tmax = fmaxf(tmax, __shfl_xor(tmax, xm, 32));
      float mn = fmaxf(mrun[r], tmax);
      float pr = __expf(sv[r] - mn);
      float corr = __expf(mrun[r] - mn);
      float ps = pr;
      for (int xm = 1; xm < 16; xm <<= 1) ps += __shfl_xor(ps, xm, 32);
      lrun[r] = lrun[r] * corr + ps;
      mrun[r] = mn;
      oacc[r] *= corr;
      sP[wv][rbase + r][mrow] = pr;
    }
    __syncthreads();

    // O += P @ V   (contraction over 16 keys -> 4 WMMAs)
    for (int kc = 0; kc < 4; ++kc) {
      v2f a, bf;
      a.x = sP[wv][mrow][kc * 4 + kb];
      a.y = sP[wv][mrow][kc * 4 + kb + 1];
      bf.x = kv[((size_t)b * N_ + j0 + kc * 4 + kb) * (2 * D_) + D_ + h * DH_ + mrow];
      bf.y = kv[((size_t)b * N_ + j0 + kc * 4 + kb + 1) * (2 * D_) + D_ + h * DH_ + mrow];
      oacc = wmma_f32(a, bf, oacc);
    }
  }

  for (int r = 0; r < 8; ++r) {
    int i = i0 + rbase + r;
    obuf[((size_t)b * N_ + i) * D_ + h * DH_ + mrow] = oacc[r] / lrun[r];
  }
}

// ---------------------------------------------------------------------------
// Depthwise causal conv(4) + bias + SiLU for the Mamba branch.
// u[b,t,d] = silu(sum_k conv_w[d,k]*xz[b,t-3+k,d] + conv_b[d])
// ---------------------------------------------------------------------------
__global__ void dwconv_silu_kernel(const float* __restrict__ xz,
                                   const float* __restrict__ cw,
                                   const float* __restrict__ cb,
                                   float* __restrict__ ub) {
  size_t tid = (size_t)blockIdx.x * blockDim.x + threadIdx.x;
  if (tid >= (size_t)B_ * N_ * DI_) return;
  int d = (int)(tid & (DI_ - 1));
  int t = (int)((tid >> 7) & (N_ - 1));
  int b = (int)(tid >> 17);
  float acc = cb[d];
  for (int k = 0; k < 4; ++k) {
    int tt = t - 3 + k;
    float xv = (tt >= 0) ? xz[((size_t)b * N_ + tt) * (2 * DI_) + d] : 0.0f;
    acc += cw[d * 4 + k] * xv;
  }
  ub[((size_t)b * N_ + t) * DI_ + d] = siluf_(acc);
}

// ---------------------------------------------------------------------------
// Mamba selective scan. One wave per (b, d): lane owns states s=lane, lane+32.
// y[b,t,d] = (sum_s h*C) + u*Dm, then gated by silu(z). Cross-lane reduce via
// wave32 shfl_xor.
// ---------------------------------------------------------------------------
__global__ __launch_bounds__(128)
void scan_kernel(const float* __restrict__ dtb, const float* __restrict__ ub,
                 const float* __restrict__ xdb, const float* __restrict__ xz,
                 const float* __restrict__ A_log, const float* __restrict__ Dm,
                 float* __restrict__ yb) {
  int wv = (int)((blockIdx.x * blockDim.x + threadIdx.x) >> 5);
  int lane = threadIdx.x & 31;
  int d = wv & (DI_ - 1);
  int b = wv >> 7;
  float A0 = -__expf(A_log[d * DS_ + lane]);
  float A1 = -__expf(A_log[d * DS_ + lane + 32]);
  float Dv = Dm[d];
  float h0 = 0.0f, h1 = 0.0f;
  for (int t = 0; t < N_; ++t) {
    size_t row = (size_t)b * N_ + t;
    if (t + 1 < N_) __builtin_prefetch(xdb + (row + 1) * 132, 0, 3);
    float dt = dtb[row * DI_ + d];
    float ut = ub[row * DI_ + d];
    const float* xr = xdb + row * 132;
    float B0 = xr[4 + lane], B1 = xr[4 + lane + 32];
    float C0 = xr[68 + lane], C1 = xr[68 + lane + 32];
    float du = dt * ut;
    h0 = h0 * __expf(dt * A0) + du * B0;
    h1 = h1 * __expf(dt * A1) + du * B1;
    float p = h0 * C0 + h1 * C1;
    for (int xm = 16; xm >= 1; xm >>= 1) p += __shfl_xor(p, xm, 32);
    if (lane == 0) {
      float y = p + ut * Dv;
      float z = xz[row * (2 * DI_) + DI_ + d];
      y *= siluf_(z);
      yb[row * DI_ + d] = y;
    }
  }
}

// ---------------------------------------------------------------------------
// RMS group norm (4 groups of 16) -> writes into zero-padded mpad at row n+3.
// One wave per row; 16-lane-half shfl reductions give per-group sums.
// ---------------------------------------------------------------------------
__global__ void rmsgn_kernel(const float* __restrict__ in,
                             const float* __restrict__ gamma,
                             float* __restrict__ mpad, int nrows) {
  int wv = (int)((blockIdx.x * blockDim.x + threadIdx.x) >> 5);
  int lane = threadIdx.x & 31;
  if (wv >= nrows) return;
  int b = wv >> 10, n = wv & (N_ - 1);
  const float* r = in + (size_t)wv * D_;
  float x0 = r[lane], x1 = r[lane + 32];
  float s0 = x0 * x0, s1 = x1 * x1;
  for (int xm = 1; xm < 16; xm <<= 1) {
    s0 += __shfl_xor(s0, xm, 32);
    s1 += __shfl_xor(s1, xm, 32);
  }
  float r0 = sqrtf(s0) * 0.25f;  // * dpg^-0.5, dpg=16
  float r1 = sqrtf(s1) * 0.25f;
  float* dst = mpad + ((size_t)b * TPAD + 3 + n) * D_;
  dst[lane]      = x0 / (r0 + 1e-5f) * gamma[lane];
  dst[lane + 32] = x1 / (r1 + 1e-5f) * gamma[lane + 32];
}

__global__ void zero_pad_kernel(float* __restrict__ mpad) {
  int tid = blockIdx.x * blockDim.x + threadIdx.x;
  if (tid >= B_ * 6 * D_) return;
  int c = tid & 63;
  int r6 = (tid >> 6) % 6;
  int b = tid / (6 * D_);
  int row = (r6 < 3) ? r6 : (TFF + r6 - 3);  // 0,1,2, 1027,1028,1029
  mpad[((size_t)b * TPAD + row) * D_ + c] = 0.0f;
}

// Pre-transpose conv weights into GEMM-friendly KxN slabs (one per tap k).
// wc1t[(k*64+ic)*512+oc] = c1_w[oc,ic,k];  wd1t[(k*256+ic)*64+oc] = d1_w[ic,oc,3-k]
__global__ void prep_weights_kernel(const float* __restrict__ c1w,
                                    const float* __restrict__ d1w,
                                    float* __restrict__ wc1t,
                                    float* __restrict__ wd1t) {
  int tid = blockIdx.x * blockDim.x + threadIdx.x;
  if (tid < 4 * 64 * 512) {
    int oc = tid & 511;
    int ic = (tid >> 9) & 63;
    int k = tid >> 15;
    wc1t[tid] = c1w[(size_t)oc * 256 + ic * 4 + k];
  } else if (tid < 4 * 64 * 512 + 4 * 256 * 64) {
    int t2 = tid - 4 * 64 * 512;
    int oc = t2 & 63;
    int ic = (t2 >> 6) & 255;
    int k = t2 >> 14;
    wd1t[t2] = d1w[((size_t)ic * 64 + oc) * 4 + (3 - k)];
  }
}

// GLU gating: g2[b,t,c] = xc2[b,t,c] * silu(xc2[b,t,256+c])
__global__ void gate_silu_kernel(const float* __restrict__ xc2,
                                 float* __restrict__ g2) {
  size_t tid = (size_t)blockIdx.x * blockDim.x + threadIdx.x;
  if (tid >= (size_t)B_ * TFF * 256) return;
  int c = (int)(tid & 255);
  size_t bt = tid >> 8;
  int t = (int)(bt % TFF);
  int b = (int)(bt / TFF);
  const float* row = xc2 + ((size_t)b * TFF + t) * 512;
  g2[((size_t)b * TFF + t) * 256 + c] = row[c] * siluf_(row[256 + c]);
}

// Final residual combine: xfin = xattn + m_rms + 0.5*(ffout + d1_b)
__global__ void combine_kernel(const float* __restrict__ xattn,
                               const float* __restrict__ mpad,
                               const float* __restrict__ ffout,
                               const float* __restrict__ d1b,
                               float* __restrict__ xfin) {
  size_t tid = (size_t)blockIdx.x * blockDim.x + threadIdx.x;
  if (tid >= (size_t)ROWS * D_) return;
  int c = (int)(tid & 63);
  size_t row = tid >> 6;
  int n = (int)(row & (N_ - 1));
  int b = (int)(row >> 10);
  float mr = mpad[((size_t)b * TPAD + 3 + n) * D_ + c];
  xfin[tid] = xattn[tid] + mr + 0.5f * (ffout[tid] + d1b[c]);
}

// ---------------------------------------------------------------------------
extern "C" void kernel_launch(void* const* d_in, const int* in_sizes, int n_in,
                              void* d_out, int out_size, void* d_ws,
                              size_t ws_size, hipStream_t stream) {
  const float* x       = (const float*)d_in[0];
  const unsigned char* drop = (const unsigned char*)d_in[1];
  const float* ln1_w   = (const float*)d_in[2];
  const float* ln1_b   = (const float*)d_in[3];
  const float* Wq      = (const float*)d_in[4];
  const float* Wkv     = (const float*)d_in[5];
  const float* Wo      = (const float*)d_in[6];
  const float* bo      = (const float*)d_in[7];
  const float* rel_emb = (const float*)d_in[8];
  const float* ln2_w   = (const float*)d_in[9];
  const float* ln2_b   = (const float*)d_in[10];
  const float* W_in    = (const float*)d_in[11];
  const float* conv_w  = (const float*)d_in[12];
  const float* conv_b  = (const float*)d_in[13];
  const float* W_xproj = (const float*)d_in[14];
  const float* W_dt    = (const float*)d_in[15];
  const float* b_dt    = (const float*)d_in[16];
  const float* A_log   = (const float*)d_in[17];
  const float* Dm      = (const float*)d_in[18];
  const float* W_out   = (const float*)d_in[19];
  const float* gamma   = (const float*)d_in[20];
  const float* c1_w    = (const float*)d_in[21];
  const float* c1_b    = (const float*)d_in[22];
  const float* d1_w    = (const float*)d_in[23];
  const float* d1_b    = (const float*)d_in[24];
  const float* ln3_w   = (const float*)d_in[25];
  const float* ln3_b   = (const float*)d_in[26];
  float* out = (float*)d_out;

  // workspace carve-up (floats)
  float* W = (float*)d_ws;
  size_t off = 0;
  auto alloc = [&](size_t n) { float* p = W + off; off += n; return p; };
  float* xt    = alloc((size_t)ROWS * D_);        // LN1 / LN2 output
  float* qbuf  = alloc((size_t)ROWS * D_);
  float* kvbuf = alloc((size_t)ROWS * 2 * D_);
  float* obuf  = alloc((size_t)ROWS * D_);        // attention output pre-Wo
  float* xattn = alloc((size_t)ROWS * D_);        // 2x + attn@Wo + bo
  float* xz    = alloc((size_t)ROWS * 2 * DI_);
  float* ubuf  = alloc((size_t)ROWS * DI_);
  float* xdb   = alloc((size_t)ROWS * 132);
  float* dtbuf = alloc((size_t)ROWS * DI_);
  float* ybuf  = alloc((size_t)ROWS * DI_);
  float* mbuf  = alloc((size_t)ROWS * D_);        // leaky(y@W_out)
  float* mpad  = alloc((size_t)B_ * TPAD * D_);   // rms-normed m, zero-padded
  float* xc2   = alloc((size_t)B_ * TFF * 512);   // c1 conv output
  float* g2    = alloc((size_t)B_ * TFF * 256);   // gated
  float* ffout = alloc((size_t)ROWS * D_);
  float* xfin  = alloc((size_t)ROWS * D_);
  float* wc1t  = alloc((size_t)4 * 64 * 512);
  float* wd1t  = alloc((size_t)4 * 256 * 64);
  (void)ws_size; (void)n_in; (void)in_sizes; (void)out_size;

  // --- attention branch ---
  ln64_kernel<<<1024, 256, 0, stream>>>(x, ln1_w, ln1_b, xt, ROWS);
  gemm16_wmma<false><<<dim3(4, 64, 1), 256, 0, stream>>>(xt, Wq, qbuf, ROWS, 64,
      64, 64, 0, 0, 64, 0, nullptr, nullptr, 0.0f, 0, 0);
  gemm16_wmma<false><<<dim3(8, 64, 1), 256, 0, stream>>>(xt, Wkv, kvbuf, ROWS,
      128, 64, 64, 0, 0, 128, 0, nullptr, nullptr, 0.0f, 0, 0);
  attn_kernel<<<B_ * H_ * 16, 128, 0, stream>>>(qbuf, kvbuf, rel_emb, drop,
                                                obuf);
  // xattn = 2*x + obuf@Wo + bo
  gemm16_wmma<false><<<dim3(4, 64, 1), 256, 0, stream>>>(obuf, Wo, xattn, ROWS,
      64, 64, 64, 0, 0, 64, 0, bo, x, 2.0f, 0, 0);

  // --- mamba branch ---
  ln64_kernel<<<1024, 256, 0, stream>>>(xattn, ln2_w, ln2_b, xt, ROWS);
  gemm16_wmma<false><<<dim3(16, 64, 1), 256, 0, stream>>>(xt, W_in, xz, ROWS,
      256, 64, 64, 0, 0, 256, 0, nullptr, nullptr, 0.0f, 0, 0);
  dwconv_silu_kernel<<<(ROWS * DI_) / 256, 256, 0, stream>>>(xz, conv_w, conv_b,
                                                             ubuf);
  gemm16_wmma<true><<<dim3(9, 64, 1), 256, 0, stream>>>(ubuf, W_xproj, xdb,
      ROWS, 132, 128, 128, 0, 0, 132, 0, nullptr, nullptr, 0.0f, 0, 0);
  // dt = softplus(xdb[:, :4] @ W_dt + b_dt)
  gemm16_wmma<false><<<dim3(8, 64, 1), 256, 0, stream>>>(xdb, W_dt, dtbuf, ROWS,
      128, 4, 132, 0, 0, 128, 0, b_dt, nullptr, 0.0f, 1, 0);
  scan_kernel<<<(B_ * DI_) / 4, 128, 0, stream>>>(dtbuf, ubuf, xdb, xz, A_log,
                                                  Dm, ybuf);
  // mbuf = leaky(y @ W_out)
  gemm16_wmma<false><<<dim3(4, 64, 1), 256, 0, stream>>>(ybuf, W_out, mbuf,
      ROWS, 64, 128, 128, 0, 0, 64, 0, nullptr, nullptr, 0.0f, 2, 0);

  // --- rms group norm into padded buffer ---
  zero_pad_kernel<<<12, 256, 0, stream>>>(mpad);
  rmsgn_kernel<<<1024, 256, 0, stream>>>(mbuf, gamma, mpad, ROWS);

  // --- FFN (conv-GLU) as shifted-A accumulating WMMA GEMMs ---
  prep_weights_kernel<<<768, 256, 0, stream>>>(c1_w, d1_w, wc1t, wd1t);
  for (int k = 0; k < 4; ++k) {  // xc2[b,t,:512] = sum_k mpad[b,t+k,:] @ Wc1_k
    gemm16_wmma<true><<<dim3(32, 9, B_), 256, 0, stream>>>(
        mpad, wc1t + (size_t)k * 64 * 512, xc2, TFF, 512, 64, 64,
        (size_t)TPAD * 64, k, 512, (size_t)TFF * 512,
        (k == 0) ? c1_b : nullptr, nullptr, 0.0f, 0, (k > 0) ? 1 : 0);
  }
  gate_silu_kernel<<<(B_ * TFF * 256 + 255) / 256, 256, 0, stream>>>(xc2, g2);
  for (int k = 0; k < 4; ++k) {  // ffout[b,n,:64] = sum_k g2[b,n+k,:] @ Wd1_k
    gemm16_wmma<false><<<dim3(4, 8, B_), 256, 0, stream>>>(
        g2, wd1t + (size_t)k * 256 * 64, ffout, N_, 64, 256, 256,
        (size_t)TFF * 256, k, 64, (size_t)N_ * 64, nullptr, nullptr, 0.0f, 0,
        (k > 0) ? 1 : 0);
  }

  // --- final combine + LN3 ---
  combine_kernel<<<(ROWS * D_) / 256, 256, 0, stream>>>(xattn, mpad, ffout,
                                                        d1_b, xfin);
  ln64_kernel<<<1024, 256, 0, stream>>>(xfin, ln3_w, ln3_b, out, ROWS);
}